// CrossAttention_21844203668060
// MI455X (gfx1250) — compile-verified
//
#include <hip/hip_runtime.h>
#include <hip/hip_bf16.h>
#include <cstdint>

// ---------------------------------------------------------------------------
// Windowed cross-attention for MI455X (gfx1250).
// fp32 WMMA (16x16x4) + Tensor Data Mover staging + fully-unrolled pipelined
// fragment loads (no loop-carried accumulator PHIs -> no v_mov_b64 copies).
// ---------------------------------------------------------------------------

typedef __attribute__((ext_vector_type(2))) float        v2f;
typedef __attribute__((ext_vector_type(8))) float        v8f;
typedef __attribute__((ext_vector_type(4))) unsigned int v4u;
typedef __attribute__((ext_vector_type(4))) int          v4i;
typedef __attribute__((ext_vector_type(8))) int          v8i;

#define BSZ    8
#define TDIM   16
#define NTOK   196
#define CDIM   512
#define HEADS  8
#define HD     64
#define QSCALE 0.125f      // 64^-0.5
#define NPAD   208         // 13 * 16
#define NT     13

__device__ __forceinline__ v8f zero8() {
  v8f z = {0.f, 0.f, 0.f, 0.f, 0.f, 0.f, 0.f, 0.f};
  return z;
}

// D = A(16x4) x B(4x16) + C, fp32 matrix core
__device__ __forceinline__ v8f wmma_f32(v2f a, v2f b, v8f c) {
  return __builtin_amdgcn_wmma_f32_16x16x4_f32(false, a, false, b, (short)0, c,
                                               false, false);
}

// ---------------------------------------------------------------------------
// Tensor Data Mover: 2D tile (fp32) global -> LDS.
// D# packing per CDNA5 ISA 8.3/8.4. dims/strides in elements (data_size=4B).
// dim1 is measured from the tile start; OOB rows are zero-filled by TDM.
// ---------------------------------------------------------------------------
__device__ __forceinline__ void tdm_load_2d_f32(unsigned int lds_off,
                                                const void* gaddr,
                                                unsigned int dim0,
                                                unsigned int dim1,
                                                unsigned int stride0,
                                                unsigned int tile0,
                                                unsigned int tile1) {
  const unsigned long long ga = (unsigned long long)(uintptr_t)gaddr;
  v4u g0;
  g0[0] = 1u;                                   // count=1, user descriptor
  g0[1] = lds_off;                              // lds_addr (bytes)
  g0[2] = (unsigned int)(ga & 0xFFFFFFFFu);     // global_addr[31:0]
  g0[3] = (unsigned int)((ga >> 32) & 0x1FFFFFFu) | (2u << 30);  // addr[56:32] | type=2
  v8i g1;
  g1[0] = (int)(2u << 16);                      // data_size=4B; wg_mask=0
  g1[1] = (int)((dim0 & 0xFFFFu) << 16);        // tensor_dim0[15:0] @ bits63:48
  g1[2] = (int)(((dim0 >> 16) & 0xFFFFu) | ((dim1 & 0xFFFFu) << 16));
  g1[3] = (int)(((dim1 >> 16) & 0xFFFFu) | ((tile0 & 0xFFFFu) << 16));
  g1[4] = (int)(tile1 & 0xFFFFu);               // tile_dim1; tile_dim2=0
  g1[5] = (int)stride0;                         // tensor_dim0_stride[31:0]
  g1[6] = 0;
  g1[7] = 0;
  v4i z4 = {0, 0, 0, 0};
  v8i z8 = {0, 0, 0, 0, 0, 0, 0, 0};
  __builtin_amdgcn_tensor_load_to_lds(g0, g1, z4, z4, z8, 0);
}

__device__ __forceinline__ unsigned int lds_offset(const void* p) {
  return (unsigned int)(uintptr_t)p;  // low 32 bits of flat-shared = LDS offset
}

// ---------------------------------------------------------------------------
// Fully-unrolled 16x64-per-wave GEMM slab over K=512.
// A staged in LDS (16x512); B = weight rows (row-major, K contiguous).
// Double-buffered b-fragments, all prefetch guards fold at compile time.
// ---------------------------------------------------------------------------
__device__ __forceinline__ void wave_gemm_512(const float (*As)[CDIM],
                                              const float* __restrict__ w,
                                              int col0, int laneM, int hi,
                                              v8f acc[4]) {
  const float* wp[4];
#pragma unroll
  for (int c = 0; c < 4; ++c)
    wp[c] = w + (size_t)(col0 + c * 16 + laneM) * CDIM + hi * 2;

  v2f b0[4], b1[4];
#pragma unroll
  for (int c = 0; c < 4; ++c) {
    b0[c] = *(const v2f*)(wp[c] + 0);
    b1[c] = *(const v2f*)(wp[c] + 4);
  }
#pragma unroll
  for (int kk = 0; kk < CDIM; kk += 8) {
    const v2f a0 = *(const v2f*)&As[laneM][kk + hi * 2];
#pragma unroll
    for (int c = 0; c < 4; ++c) acc[c] = wmma_f32(a0, b0[c], acc[c]);
    if (kk + 8 < CDIM) {
#pragma unroll
      for (int c = 0; c < 4; ++c) b0[c] = *(const v2f*)(wp[c] + kk + 8);
    }
    const v2f a1 = *(const v2f*)&As[laneM][kk + 4 + hi * 2];
#pragma unroll
    for (int c = 0; c < 4; ++c) acc[c] = wmma_f32(a1, b1[c], acc[c]);
    if (kk + 12 < CDIM) {
#pragma unroll
      for (int c = 0; c < 4; ++c) b1[c] = *(const v2f*)(wp[c] + kk + 12);
    }
  }
}

// ---------------------------------------------------------------------------
// Kernel 1: qkv = x @ w_qkv^T  (25088 x 512 x 1536), scatter to q/k/v
// grid = (1568, 3), block = 256 (8 waves); wave covers a 16x64 strip.
// ---------------------------------------------------------------------------
__global__ __launch_bounds__(256) void qkv_kernel(
    const float* __restrict__ x, const float* __restrict__ w,
    float* __restrict__ qb, float* __restrict__ kb, float* __restrict__ vb) {
  __shared__ float As[16][CDIM];  // 32 KB, staged by TDM
  const int tid = threadIdx.x;
  const int m0  = blockIdx.x * 16;

  if (tid < 32) {  // wave 0 drives the tensor DMA
    tdm_load_2d_f32(lds_offset(&As[0][0]), x + (size_t)m0 * CDIM,
                    /*dim0=*/CDIM, /*dim1=*/25088 - m0, /*stride0=*/CDIM,
                    /*tile0=*/CDIM, /*tile1=*/16);
    __builtin_amdgcn_s_wait_tensorcnt(0);
  }
  __syncthreads();

  const int wave = tid >> 5, lane = tid & 31;
  const int laneM = lane & 15, hi = lane >> 4;
  const int col0 = blockIdx.y * 512 + wave * 64;

  v8f acc[4] = {zero8(), zero8(), zero8(), zero8()};
  wave_gemm_512(As, w, col0, laneM, hi, acc);

#pragma unroll
  for (int c = 0; c < 4; ++c) {
    const int col   = col0 + c * 16 + laneM;
    const int which = col >> 9;           // 0=q 1=k 2=v
    const int h     = (col >> 6) & 7;
    const int dd    = col & 63;
    float* dst = (which == 0) ? qb : ((which == 1) ? kb : vb);
    const float sc = (which == 0) ? QSCALE : 1.0f;
#pragma unroll
    for (int r = 0; r < 8; ++r) {
      const int rg = m0 + r + hi * 8;     // flat row over (b*T, n)
      const int bt = rg / NTOK;
      const int n  = rg - bt * NTOK;
      const int bb = bt >> 4, tt = bt & 15;
      const size_t off =
          ((size_t)(((bb * HEADS + h) * TDIM + tt) * NTOK + n)) * HD + dd;
      dst[off] = acc[c][r] * sc;
    }
  }
}

// ---------------------------------------------------------------------------
// Kernel 2: attention, one (b,h,s) per block; block = 128 (4 waves).
//   scores(n,m) = q[s-w*(n,m), n] . k[s, m]  (last-window-wins select)
//   out = softmax(scores) @ v[s], stored with the reference's head-mixing
//   flat layout: ob[bt*196*512 + h*12544 + n*64 + dd]
// Per-wave LDS region (3328 floats) is used twice per row-tile:
//   phase A: 3 q time-slices (16x64 each) staged by TDM (OOB rows -> 0)
//   phase B: softmax'd P strip (16x208) for A-fragment reload in P@V
// ---------------------------------------------------------------------------
__global__ __launch_bounds__(128) void attn_kernel(
    const float* __restrict__ q, const float* __restrict__ k,
    const float* __restrict__ v, float* __restrict__ ob) {
  __shared__ float wsm[4 * 16 * NPAD];  // 53248 B

  const int tid  = threadIdx.x;
  const int wave = tid >> 5, lane = tid & 31;
  const int laneM = lane & 15, hi = lane >> 4;

  const int gid = blockIdx.x;
  const int s = gid & 15;
  const int h = (gid >> 4) & 7;
  const int b = gid >> 7;

  const size_t bh  = (size_t)b * HEADS + h;
  const float* qbh = q + bh * (TDIM * NTOK * HD);
  const float* qs0 = qbh + (size_t)s * (NTOK * HD);
  const float* qs1 = qbh + (size_t)(s >= 1 ? s - 1 : 0) * (NTOK * HD);
  const float* qs2 = qbh + (size_t)(s >= 2 ? s - 2 : 0) * (NTOK * HD);
  const float* kp  = k + (bh * TDIM + s) * (NTOK * HD);
  const float* vp  = v + (bh * TDIM + s) * (NTOK * HD);

  float* pw = wsm + wave * 16 * NPAD;
  const unsigned int pw_off = lds_offset(pw);

  for (int rt = wave; rt < NT; rt += 4) {
    const int n0 = rt * 16;

    // ---- phase A: TDM-stage q[s], q[s-1], q[s-2] rows n0..n0+15 ----
    tdm_load_2d_f32(pw_off + 0u,    qs0 + (size_t)n0 * HD, HD,
                    (unsigned)(NTOK - n0), HD, HD, 16);
    tdm_load_2d_f32(pw_off + 4096u, qs1 + (size_t)n0 * HD, HD,
                    (unsigned)(NTOK - n0), HD, HD, 16);
    tdm_load_2d_f32(pw_off + 8192u, qs2 + (size_t)n0 * HD, HD,
                    (unsigned)(NTOK - n0), HD, HD, 16);
    __builtin_amdgcn_s_wait_tensorcnt(0);

    const float* q0l = pw + 0    + laneM * HD + hi * 2;
    const float* q1l = pw + 1024 + laneM * HD + hi * 2;
    const float* q2l = pw + 2048 + laneM * HD + hi * 2;

    // ---- scores: 13 m-tiles, 3 q-times each, pipelined k fragments ----
    v8f p[NT];
#pragma unroll
    for (int mt = 0; mt < NT; ++mt) {
      const int m0   = mt * 16;
      const int mcol = (m0 + laneM < NTOK) ? (m0 + laneM) : (NTOK - 1);
      const float* kr = kp + (size_t)mcol * HD + hi * 2;

      v8f a0 = zero8(), a1 = zero8(), a2 = zero8();
      v2f fb0 = *(const v2f*)(kr + 0);
      v2f fb1 = *(const v2f*)(kr + 4);
#pragma unroll
      for (int kk = 0; kk < HD; kk += 8) {
        a0 = wmma_f32(*(const v2f*)(q0l + kk), fb0, a0);
        a1 = wmma_f32(*(const v2f*)(q1l + kk), fb0, a1);
        a2 = wmma_f32(*(const v2f*)(q2l + kk), fb0, a2);
        if (kk + 8 < HD) fb0 = *(const v2f*)(kr + kk + 8);
        a0 = wmma_f32(*(const v2f*)(q0l + kk + 4), fb1, a0);
        a1 = wmma_f32(*(const v2f*)(q1l + kk + 4), fb1, a1);
        a2 = wmma_f32(*(const v2f*)(q2l + kk + 4), fb1, a2);
        if (kk + 12 < HD) fb1 = *(const v2f*)(kr + kk + 12);
      }

      // last-window-wins selection (w* in {0,1,2})
      const int  m   = m0 + laneM;
      const bool c2m = (s >= 3) && (m >= 28);
      const bool c1m = (s >= 2) && (m >= 14);
      v8f sel;
#pragma unroll
      for (int r = 0; r < 8; ++r) {
        const int n = n0 + r + hi * 8;
        float val = (c2m && n <= 167) ? a2[r]
                  : ((c1m && n <= 181) ? a1[r] : a0[r]);
        if (m >= NTOK) val = -1e30f;  // pad cols -> exp() == 0
        sel[r] = val;
      }
      p[mt] = sel;
    }

    // ---- softmax over m (rows across 16-lane halves, shfl_xor reduce) ----
#pragma unroll
    for (int r = 0; r < 8; ++r) {
      float mx = -1e30f;
#pragma unroll
      for (int mt = 0; mt < NT; ++mt) mx = fmaxf(mx, p[mt][r]);
#pragma unroll
      for (int off = 1; off < 16; off <<= 1)
        mx = fmaxf(mx, __shfl_xor(mx, off, 32));
      float sm = 0.f;
#pragma unroll
      for (int mt = 0; mt < NT; ++mt) {
        const float e = __expf(p[mt][r] - mx);
        p[mt][r] = e;
        sm += e;
      }
#pragma unroll
      for (int off = 1; off < 16; off <<= 1) sm += __shfl_xor(sm, off, 32);
      const float inv = 1.0f / sm;
#pragma unroll
      for (int mt = 0; mt < NT; ++mt) p[mt][r] *= inv;
    }

    // ---- phase B: stash P (C-layout) to LDS for A-fragment reload ----
#pragma unroll
    for (int mt = 0; mt < NT; ++mt)
#pragma unroll
      for (int r = 0; r < 8; ++r)
        pw[(r + hi * 8) * NPAD + mt * 16 + laneM] = p[mt][r];

    // ---- out = P(16x208) @ V(208x64), pipelined v fragments ----
    const float* pl = pw + laneM * NPAD + hi * 2;
    v8f o[4] = {zero8(), zero8(), zero8(), zero8()};
    v2f vf0[4], vf1[4];
    {
      const int ka0 = hi * 2, ka1 = 4 + hi * 2;
#pragma unroll
      for (int c = 0; c < 4; ++c) {
        v2f t0, t1;
        t0.x = vp[(size_t)ka0 * HD + c * 16 + laneM];
        t0.y = vp[(size_t)(ka0 + 1) * HD + c * 16 + laneM];
        t1.x = vp[(size_t)ka1 * HD + c * 16 + laneM];
        t1.y = vp[(size_t)(ka1 + 1) * HD + c * 16 + laneM];
        vf0[c] = t0;
        vf1[c] = t1;
      }
    }
#pragma unroll
    for (int kk = 0; kk < NPAD; kk += 8) {
      {
        const v2f a = *(const v2f*)(pl + kk);
#pragma unroll
        for (int c = 0; c < 4; ++c) o[c] = wmma_f32(a, vf0[c], o[c]);
      }
      if (kk + 8 < NPAD) {
        const int ka = kk + 8 + hi * 2;
        const int r0 = (ka < NTOK) ? ka : NTOK - 1;
        const int r1 = (ka + 1 < NTOK) ? ka + 1 : NTOK - 1;
#pragma unroll
        for (int c = 0; c < 4; ++c) {
          v2f t;
          t.x = vp[(size_t)r0 * HD + c * 16 + laneM];
          t.y = vp[(size_t)r1 * HD + c * 16 + laneM];
          vf0[c] = t;
        }
      }
      {
        const v2f a = *(const v2f*)(pl + kk + 4);
#pragma unroll
        for (int c = 0; c < 4; ++c) o[c] = wmma_f32(a, vf1[c], o[c]);
      }
      if (kk + 12 < NPAD) {
        const int ka = kk + 12 + hi * 2;
        const int r0 = (ka < NTOK) ? ka : NTOK - 1;
        const int r1 = (ka + 1 < NTOK) ? ka + 1 : NTOK - 1;
#pragma unroll
        for (int c = 0; c < 4; ++c) {
          v2f t;
          t.x = vp[(size_t)r0 * HD + c * 16 + laneM];
          t.y = vp[(size_t)r1 * HD + c * 16 + laneM];
          vf1[c] = t;
        }
      }
    }

    // ---- store with the reference's head-mixing flat layout ----
    const size_t btBase = (size_t)(b * TDIM + s) * (NTOK * CDIM);
#pragma unroll
    for (int c = 0; c < 4; ++c) {
#pragma unroll
      for (int r = 0; r < 8; ++r) {
        const int n = n0 + r + hi * 8;
        if (n < NTOK) {
          const size_t off =
              btBase + (size_t)h * (NTOK * HD) + (size_t)n * HD + c * 16 + laneM;
          ob[off] = o[c][r];
        }
      }
    }
  }
}

// ---------------------------------------------------------------------------
// Kernel 3: out = A(25088x512) @ w_proj^T(512x512) + b_proj
// grid = 1568, block = 256 (8 waves x 64 cols = 512 cols)
// ---------------------------------------------------------------------------
__global__ __launch_bounds__(256) void proj_kernel(
    const float* __restrict__ a_in, const float* __restrict__ w,
    const float* __restrict__ bias, float* __restrict__ out) {
  __shared__ float As[16][CDIM];
  const int tid = threadIdx.x;
  const int m0  = blockIdx.x * 16;

  if (tid < 32) {
    tdm_load_2d_f32(lds_offset(&As[0][0]), a_in + (size_t)m0 * CDIM,
                    CDIM, 25088 - m0, CDIM, CDIM, 16);
    __builtin_amdgcn_s_wait_tensorcnt(0);
  }
  __syncthreads();

  const int wave = tid >> 5, lane = tid & 31;
  const int laneM = lane & 15, hi = lane >> 4;
  const int col0 = wave * 64;

  v8f acc[4] = {zero8(), zero8(), zero8(), zero8()};
  wave_gemm_512(As, w, col0, laneM, hi, acc);

#pragma unroll
  for (int c = 0; c < 4; ++c) {
    const int col = col0 + c * 16 + laneM;
    const float bb = bias[col];
#pragma unroll
    for (int r = 0; r < 8; ++r) {
      const int rg = m0 + r + hi * 8;
      out[(size_t)rg * CDIM + col] = acc[c][r] + bb;
    }
  }
}

// ---------------------------------------------------------------------------
extern "C" void kernel_launch(void* const* d_in, const int* in_sizes, int n_in,
                              void* d_out, int out_size, void* d_ws,
                              size_t ws_size, hipStream_t stream) {
  (void)in_sizes; (void)n_in; (void)out_size; (void)ws_size;
  const float* x      = (const float*)d_in[0];
  const float* w_qkv  = (const float*)d_in[1];
  const float* w_proj = (const float*)d_in[2];
  const float* b_proj = (const float*)d_in[3];
  // d_in[4] = b_size (constant 8, baked in)

  float* ws = (float*)d_ws;
  const size_t QSZ = (size_t)BSZ * HEADS * TDIM * NTOK * HD;  // 12,845,056 floats
  float* qb = ws;
  float* kb = ws + QSZ;
  float* vb = ws + 2 * QSZ;
  float* ob = ws + 3 * QSZ;

  qkv_kernel<<<dim3(25088 / 16, 3), 256, 0, stream>>>(x, w_qkv, qb, kb, vb);
  attn_kernel<<<dim3(BSZ * HEADS * TDIM), 128, 0, stream>>>(qb, kb, vb, ob);
  proj_kernel<<<dim3(25088 / 16), 256, 0, stream>>>(ob, w_proj, b_proj,
                                                    (float*)d_out);
}